// LSHDecoder_23716809408540
// MI455X (gfx1250) — compile-verified
//
#include <hip/hip_runtime.h>
#include <hip/hip_bf16.h>
#include <stdint.h>

typedef __attribute__((ext_vector_type(2))) float v2f;
typedef __attribute__((ext_vector_type(8))) float v8f;

#define D 256
#define TILE_M 128
#define TILE_N 64

// ---------------------------------------------------------------------------
// Kernel 1: inverse norms. One wave (32 lanes) per row of Z.
// ---------------------------------------------------------------------------
__global__ __launch_bounds__(256) void lsh_norm_kernel(const float* __restrict__ Z,
                                                       float* __restrict__ invn, int N) {
    int row  = (blockIdx.x * blockDim.x + threadIdx.x) >> 5;
    int lane = threadIdx.x & 31;
    if (row >= N) return;
    const float* z = Z + (size_t)row * D;
    float s = 0.0f;
#pragma unroll
    for (int d = lane; d < D; d += 32) {
        float x = z[d];
        s += x * x;
    }
#pragma unroll
    for (int off = 16; off > 0; off >>= 1) s += __shfl_xor(s, off, 32);
    if (lane == 0) invn[row] = 1.0f / sqrtf(s);
}

// ---------------------------------------------------------------------------
// Kernel 2: packed band codes. One 128-thread block per node.
// Thread t dots hyperplane t with Z[n]; wave32 ballot packs 4 one-byte band
// codes per wave (lane = band_in_wave*8 + bit, bit weight 2^bit as in the
// reference einsum). pcodes[n] = uint4 = all 16 band bytes.
// ---------------------------------------------------------------------------
__global__ __launch_bounds__(128) void lsh_codes_kernel(const float* __restrict__ Z,
                                                        const float* __restrict__ planes,
                                                        uint32_t* __restrict__ pcodes, int N) {
    __shared__ float zs[D];
    int n = blockIdx.x;
    const float* z = Z + (size_t)n * D;
    for (int d = threadIdx.x; d < D; d += 128) zs[d] = z[d];
    __syncthreads();

    const float4* p4 = (const float4*)(planes + (size_t)threadIdx.x * D);
    const float4* z4 = (const float4*)zs;
    float s = 0.0f;
#pragma unroll 4
    for (int d = 0; d < D / 4; ++d) {
        float4 a = p4[d];
        float4 b = z4[d];
        s += a.x * b.x + a.y * b.y + a.z * b.z + a.w * b.w;
    }
    unsigned long long bal = __ballot(s >= 0.0f);   // wave32: low 32 bits valid
    if ((threadIdx.x & 31) == 0)
        pcodes[(size_t)n * 4 + (threadIdx.x >> 5)] = (uint32_t)bal;
}

// any byte of x equal to zero?
__device__ __forceinline__ bool haszero(uint32_t x) {
    return ((x - 0x01010101u) & ~x & 0x80808080u) != 0u;
}

// ---------------------------------------------------------------------------
// Kernel 3: masked cosine-similarity GEMM with FP32 WMMA.
// 8 waves/block, 128x64 tile, each wave a 32x32 subtile (2x2 16x16 accums).
// A fragment (16x4 f32): lane l holds row (l&15), K pair 2*(l>>4)   -> float2
// B fragment (4x16 f32): lane l holds col (l&15), K pair 2*(l>>4)   -> float2
// C/D (16x16 f32): VGPR v, lanes 0-15 -> (M=v, N=lane); 16-31 -> (M=v+8).
// ---------------------------------------------------------------------------
__global__ __launch_bounds__(256) void lsh_sim_kernel(const float* __restrict__ Z,
                                                      const float* __restrict__ invn,
                                                      const uint32_t* __restrict__ pcodes,
                                                      float* __restrict__ out, int N) {
    __shared__ float    s_inr[TILE_M];
    __shared__ float    s_inc[TILE_N];
    __shared__ uint32_t s_pr[TILE_M * 4];
    __shared__ uint32_t s_pc[TILE_N * 4];

    const int i0 = blockIdx.x * TILE_M;
    const int j0 = blockIdx.y * TILE_N;
    const int tid = threadIdx.x;

    // Stage per-row/col mask metadata into LDS (all L2-resident, tiny).
    for (int t = tid; t < TILE_M; t += 256) s_inr[t] = invn[i0 + t];
    for (int t = tid; t < TILE_N; t += 256) s_inc[t] = invn[j0 + t];
    for (int t = tid; t < TILE_M * 4; t += 256) s_pr[t] = pcodes[(size_t)i0 * 4 + t];
    for (int t = tid; t < TILE_N * 4; t += 256) s_pc[t] = pcodes[(size_t)j0 * 4 + t];
    __syncthreads();

    const int wave = tid >> 5;
    const int lane = tid & 31;
    const int wm = wave >> 1;          // 4 wave-rows
    const int wn = wave & 1;           // 2 wave-cols
    const int wi = i0 + wm * 32;
    const int wj = j0 + wn * 32;

    const int r  = lane & 15;          // row/col within 16-fragment
    const int ks = (lane >> 4) << 1;   // K pair selected by lane half

    const float* a0p = Z + (size_t)(wi + r)      * D + ks;
    const float* a1p = Z + (size_t)(wi + 16 + r) * D + ks;
    const float* b0p = Z + (size_t)(wj + r)      * D + ks;
    const float* b1p = Z + (size_t)(wj + 16 + r) * D + ks;

    v8f acc00 = {}, acc01 = {}, acc10 = {}, acc11 = {};

#pragma unroll 4
    for (int k = 0; k < D; k += 4) {
        v2f a0 = *(const v2f*)(a0p + k);
        v2f a1 = *(const v2f*)(a1p + k);
        v2f b0 = *(const v2f*)(b0p + k);
        v2f b1 = *(const v2f*)(b1p + k);
        acc00 = __builtin_amdgcn_wmma_f32_16x16x4_f32(false, a0, false, b0, (short)0, acc00, false, false);
        acc01 = __builtin_amdgcn_wmma_f32_16x16x4_f32(false, a0, false, b1, (short)0, acc01, false, false);
        acc10 = __builtin_amdgcn_wmma_f32_16x16x4_f32(false, a1, false, b0, (short)0, acc10, false, false);
        acc11 = __builtin_amdgcn_wmma_f32_16x16x4_f32(false, a1, false, b1, (short)0, acc11, false, false);
    }

    // Epilogue: normalize, candidate-mask via packed-code byte equality,
    // threshold, store. Every element of the 128x64 tile is written.
    const int lhi = lane >> 4;
    v8f accs[2][2] = {{acc00, acc01}, {acc10, acc11}};

#pragma unroll
    for (int mi = 0; mi < 2; ++mi) {
#pragma unroll
        for (int ni = 0; ni < 2; ++ni) {
            const int colL = wn * 32 + ni * 16 + r;
            const int col  = j0 + colL;
            const float inc = s_inc[colL];
            const uint32_t c0 = s_pc[colL * 4 + 0];
            const uint32_t c1 = s_pc[colL * 4 + 1];
            const uint32_t c2 = s_pc[colL * 4 + 2];
            const uint32_t c3 = s_pc[colL * 4 + 3];
            v8f A = accs[mi][ni];
#pragma unroll
            for (int v = 0; v < 8; ++v) {
                const int rowL = wm * 32 + mi * 16 + v + 8 * lhi;
                const int row  = i0 + rowL;
                float val = A[v] * s_inr[rowL] * inc;
                bool cand = haszero(s_pr[rowL * 4 + 0] ^ c0) |
                            haszero(s_pr[rowL * 4 + 1] ^ c1) |
                            haszero(s_pr[rowL * 4 + 2] ^ c2) |
                            haszero(s_pr[rowL * 4 + 3] ^ c3);
                float o = (cand && (row != col) && (val > 0.5f)) ? val : 0.0f;
                out[(size_t)row * N + col] = o;
            }
        }
    }
}

// ---------------------------------------------------------------------------
extern "C" void kernel_launch(void* const* d_in, const int* in_sizes, int n_in,
                              void* d_out, int out_size, void* d_ws, size_t ws_size,
                              hipStream_t stream) {
    const float* Z      = (const float*)d_in[0];
    const float* planes = (const float*)d_in[1];
    float* out          = (float*)d_out;

    const int N = in_sizes[0] / D;   // 8192

    float*    invn   = (float*)d_ws;                       // N floats
    uint32_t* pcodes = (uint32_t*)((char*)d_ws + (size_t)N * sizeof(float)); // N * uint4

    // 1 wave per row -> 8 rows per 256-thread block
    lsh_norm_kernel<<<(N + 7) / 8, 256, 0, stream>>>(Z, invn, N);
    // 1 block (128 threads == 128 hyperplanes) per node
    lsh_codes_kernel<<<N, 128, 0, stream>>>(Z, planes, pcodes, N);
    // masked similarity GEMM
    dim3 grid(N / TILE_M, N / TILE_N);
    lsh_sim_kernel<<<grid, 256, 0, stream>>>(Z, invn, pcodes, out, N);
}